// GroupFPS_6511170420988
// MI455X (gfx1250) — compile-verified
//
#include <hip/hip_runtime.h>
#include <hip/hip_bf16.h>

#define B_ 32
#define N_ 16384
#define G_ 128
#define K_ 32
#define DSTRIDE 2052   // 2048 + 4 pad: shifts banks per row so the two WMMA halves don't collide

typedef float v2f __attribute__((ext_vector_type(2)));
typedef float v8f __attribute__((ext_vector_type(8)));
typedef unsigned int v4u __attribute__((ext_vector_type(4)));
typedef int v8i __attribute__((ext_vector_type(8)));
typedef int v4i __attribute__((ext_vector_type(4)));

// ---------- wave32 reductions (tie-break toward smaller index = jnp argmax/top_k stability) ----------
__device__ __forceinline__ void wave_argmax_f(float &v, int &i) {
#pragma unroll
  for (int off = 16; off > 0; off >>= 1) {
    float ov = __shfl_xor(v, off, 32);
    int   oi = __shfl_xor(i, off, 32);
    if (ov > v || (ov == v && oi < i)) { v = ov; i = oi; }
  }
}

__device__ __forceinline__ void wave_argmin_f(float &v, int &key, int &payload) {
#pragma unroll
  for (int off = 16; off > 0; off >>= 1) {
    float ov = __shfl_xor(v, off, 32);
    int   ok = __shfl_xor(key, off, 32);
    int   op = __shfl_xor(payload, off, 32);
    if (ov < v || (ov == v && ok < key)) { v = ov; key = ok; payload = op; }
  }
}

// ---------- Tensor Data Mover: 1-D DMA of `nelem` dwords from global to LDS ----------
// D# built per CDNA5 ISA (count=1, type=2 "image",
// data_size=4B, tile_dim0 = tensor_dim0 = nelem, 1-D: higher dims unused/zero).
// This toolchain exposes the 6-arg clang-23 builtin:
//   (uint32x4 g0, int32x8 g1, int32x4 g2, int32x4 g3, int32x8 g4, i32 cpol)
__device__ __forceinline__ void tdm_load_dwords_to_lds(void* lds_ptr, const void* gptr,
                                                       unsigned int nelem) {
  unsigned int lds_addr = (unsigned int)(unsigned long long)lds_ptr;  // LDS offset = low 32 bits
  unsigned long long ga = (unsigned long long)gptr;

  v4u g0;
  g0.x = 1u;                                            // count=1, is_restore=0, gather=0
  g0.y = lds_addr;                                      // lds_addr [63:32]
  g0.z = (unsigned int)ga;                              // global_addr [95:64]
  g0.w = (unsigned int)((ga >> 32) & 0x01FFFFFFu)       // global_addr [120:96]
       | (2u << 30);                                    // type=2 (bits 127:126)

  v8i g1;
  g1[0] = (int)(2u << 16);                              // workgroup_mask=0, data_size=2 (4B)
  g1[1] = (int)((nelem & 0xFFFFu) << 16);               // tensor_dim0[15:0] in [63:48]
  g1[2] = (int)((nelem >> 16) & 0xFFFFu);               // tensor_dim0[31:16] in [79:64]
  g1[3] = (int)((nelem & 0xFFFFu) << 16);               // tile_dim0 in [127:112]
  g1[4] = 0;                                            // tile_dim1/2 unused
  g1[5] = (int)nelem;                                   // tensor_dim0_stride[31:0]
  g1[6] = 0;
  g1[7] = 0;

  v4i gz4 = {0, 0, 0, 0};                               // groups 2/3 unused (<=2D)
  v8i gz8 = {0, 0, 0, 0, 0, 0, 0, 0};                   // extra group (clang-23 form)
  __builtin_amdgcn_tensor_load_to_lds(g0, g1, gz4, gz4, gz8, 0 /*cpol*/);
}

// =====================================================================================
// Kernel 1: Farthest-point sampling. One block per batch. All 16384 points (interleaved
// xyz, TDM-DMA'd into LDS in one tensor_load_to_lds) + the 64KB running min-distance
// array live in the 320KB WGP LDS. 1024 threads = 32 wave32s.
// =====================================================================================
__global__ __launch_bounds__(1024) void fps_kernel(const float* __restrict__ x,
                                                   float* __restrict__ cout,
                                                   float4* __restrict__ cws) {
  extern __shared__ float smem[];
  float* xyz = smem;            // [3*N] interleaved (stride-3 dword access: coprime with 64 banks)
  float* md  = xyz + 3 * N_;    // [N] running min squared distance
  __shared__ float rv[32];
  __shared__ int   ri[32];
  __shared__ int   bsel;

  const int b    = blockIdx.x;
  const int tid  = threadIdx.x;
  const int lane = tid & 31;
  const int wv   = tid >> 5;
  const float* xb = x + (size_t)b * N_ * 3;

  // Single TDM DMA: 192KB of points HBM -> LDS, no VGPR round-trip.
  if (wv == 0) {
    tdm_load_dwords_to_lds(xyz, xb, 3u * N_);
    __builtin_amdgcn_s_wait_tensorcnt(0);
  }
  __syncthreads();

  // min_d = dist^2 to point 0; center 0 = point 0
  float x0 = xyz[0], y0 = xyz[1], z0 = xyz[2];
  for (int j = 0; j < N_ / 1024; ++j) {
    int i = tid + j * 1024;
    float dx = xyz[3 * i] - x0, dy = xyz[3 * i + 1] - y0, dz = xyz[3 * i + 2] - z0;
    md[i] = dx * dx + dy * dy + dz * dz;
  }
  if (tid == 0) {
    size_t o = ((size_t)b * G_) * 3;
    cout[o + 0] = x0; cout[o + 1] = y0; cout[o + 2] = z0;
    cws[b * G_] = make_float4(x0, y0, z0, 0.f);
  }
  __syncthreads();

  for (int g = 1; g < G_; ++g) {
    // block-wide argmax over md (first-max = lowest index, like jnp.argmax)
    float bv = -1.0f; int bi = tid;
#pragma unroll
    for (int j = 0; j < N_ / 1024; ++j) {
      int i = tid + j * 1024;            // interleaved -> conflict-free LDS banks
      float v = md[i];
      if (v > bv) { bv = v; bi = i; }    // ascending i + strict '>' keeps lowest index
    }
    wave_argmax_f(bv, bi);
    if (lane == 0) { rv[wv] = bv; ri[wv] = bi; }
    __syncthreads();
    if (wv == 0) {
      float v2 = rv[lane]; int i2 = ri[lane];
      wave_argmax_f(v2, i2);
      if (lane == 0) bsel = i2;
    }
    __syncthreads();
    int sel = bsel;
    float sx = xyz[3 * sel], sy = xyz[3 * sel + 1], sz = xyz[3 * sel + 2];
    if (tid == 0) {
      size_t o = ((size_t)b * G_ + g) * 3;
      cout[o + 0] = sx; cout[o + 1] = sy; cout[o + 2] = sz;
      cws[b * G_ + g] = make_float4(sx, sy, sz, 0.f);
    }
#pragma unroll
    for (int j = 0; j < N_ / 1024; ++j) {
      int i = tid + j * 1024;
      float dx = xyz[3 * i] - sx, dy = xyz[3 * i + 1] - sy, dz = xyz[3 * i + 2] - sz;
      md[i] = fminf(md[i], dx * dx + dy * dy + dz * dz);
    }
    __syncthreads();
  }
}

// =====================================================================================
// Kernel 2: KNN via V_WMMA_F32_16X16X4_F32 distance tiles + LDS-chunked top-32 selection.
// Grid: 32 batches x 8 center-tiles. Block: 256 threads = 8 wave32s, 16 centers.
// d2' = ||x||^2 - 2 c.x  (drop per-row ||c||^2: cannot change per-row top-k order)
//   A[m][:] = [-2cx, -2cy, -2cz, 1]   B[:][n] = [x, y, z, ||x||^2]
// =====================================================================================
__global__ __launch_bounds__(256) void knn_kernel(const float* __restrict__ x,
                                                  const float4* __restrict__ cws,
                                                  float* __restrict__ pout,
                                                  int* __restrict__ iout) {
  extern __shared__ float smem[];
  float* dist  = smem;                       // [16][DSTRIDE]
  float* candD = dist + 16 * DSTRIDE;        // [16][256]
  int*   candI = (int*)(candD + 16 * 256);   // [16][256]

  const int b     = blockIdx.x >> 3;
  const int ctile = blockIdx.x & 7;
  const int tid   = threadIdx.x;
  const int lane  = tid & 31;
  const int wv    = tid >> 5;
  const int hh    = lane >> 4;               // half-wave: 0 => K0/K1, 1 => K2/K3
  const int l15   = lane & 15;
  const float* xb = x + (size_t)b * N_ * 3;
  const float INF = 1e30f;

  // A fragment (16x4 f32): lanes 0-15 hold M=0..15 K={0,1}; lanes 16-31 hold K={2,3}
  float4 c4 = cws[b * G_ + ctile * 16 + l15];
  v2f a;
  a.x = hh ? (-2.0f * c4.z) : (-2.0f * c4.x);
  a.y = hh ? 1.0f           : (-2.0f * c4.y);

  for (int chunk = 0; chunk < 8; ++chunk) {
    const int cb = chunk * 2048;
    // ---- WMMA distance tiles: each wave computes 16 tiles of 16 points ----
    for (int t = 0; t < 16; ++t) {
      int nl = (wv * 16 + t) * 16 + l15;                 // column 0..2047 inside chunk
      const float* pp = xb + (size_t)(cb + nl) * 3;
      float xx = pp[0], yy = pp[1], zz = pp[2];
      float w  = xx * xx + yy * yy + zz * zz;
      v2f bf;                                            // B 4x16: mirrors A's half-wave K split
      bf.x = hh ? zz : xx;
      bf.y = hh ? w  : yy;
      if (t + 1 < 16) __builtin_prefetch(pp + 48, 0, 1); // global_prefetch_b8 for next tile
      v8f cz = {};
      v8f d = __builtin_amdgcn_wmma_f32_16x16x4_f32(
          /*neg_a=*/false, a, /*neg_b=*/false, bf,
          /*c_mod=*/(short)0, cz, /*reuse_a=*/false, /*reuse_b=*/false);
#pragma unroll
      for (int r = 0; r < 8; ++r) {                      // D: lanes 0-15 rows 0-7, lanes 16-31 rows 8-15
        int m = r + (hh << 3);
        dist[m * DSTRIDE + nl] = d[r];
      }
    }
    __syncthreads();
    // ---- per-chunk top-32 per center: 2 centers per wave, 32 argmin rounds ----
    for (int which = 0; which < 2; ++which) {
      int m = wv * 2 + which;
      float* drow = dist + m * DSTRIDE;
      for (int k = 0; k < K_; ++k) {
        float bv = INF; int bn = 0x7fffffff; int dummy = 0;
        for (int j = 0; j < 64; ++j) {
          int n = j * 32 + lane;                         // consecutive lanes -> consecutive banks
          float v = drow[n];
          if (v < bv) { bv = v; bn = n; }
        }
        wave_argmin_f(bv, bn, dummy);
        drow[bn] = INF;                                  // all lanes, same addr/value: per-lane DS order safe
        if (lane == 0) {
          candD[m * 256 + chunk * K_ + k] = bv;
          candI[m * 256 + chunk * K_ + k] = cb + bn;
        }
      }
    }
    __syncthreads();
  }

  // ---- final: top-32 of 256 candidates per center, emitted in ascending order ----
  for (int which = 0; which < 2; ++which) {
    int m  = wv * 2 + which;                             // cand rows were written by this same wave
    int mg = ctile * 16 + m;
    for (int k = 0; k < K_; ++k) {
      float bv = INF; int bg = 0x7fffffff; int bslot = 0;
      for (int j = 0; j < 8; ++j) {
        int s = j * 32 + lane;
        float v = candD[m * 256 + s];
        int gi  = candI[m * 256 + s];
        if (v < bv || (v == bv && gi < bg)) { bv = v; bg = gi; bslot = s; }
      }
      wave_argmin_f(bv, bg, bslot);
      candD[m * 256 + bslot] = INF;                      // all lanes write same value
      if (lane == 0) {
        size_t o = ((size_t)b * G_ + mg) * K_ + k;
        iout[o] = bg;
        const float* pp = xb + (size_t)bg * 3;
        pout[o * 3 + 0] = pp[0];
        pout[o * 3 + 1] = pp[1];
        pout[o * 3 + 2] = pp[2];
      }
    }
  }
}

extern "C" void kernel_launch(void* const* d_in, const int* in_sizes, int n_in,
                              void* d_out, int out_size, void* d_ws, size_t ws_size,
                              hipStream_t stream) {
  (void)in_sizes; (void)n_in; (void)out_size; (void)ws_size;
  const float* x = (const float*)d_in[0];

  // Outputs concatenated flat: p [B,G,K,3] f32, c [B,G,3] f32, i [B,G,K] i32
  float* out  = (float*)d_out;
  float* pout = out;
  float* cout = out + (size_t)B_ * G_ * K_ * 3;
  int*   iout = (int*)(out + (size_t)B_ * G_ * K_ * 3 + (size_t)B_ * G_ * 3);
  float4* cws = (float4*)d_ws;   // FPS centers staged for kernel 2

  size_t fps_lds = (size_t)4 * N_ * sizeof(float);                       // 256 KB
  fps_kernel<<<B_, 1024, fps_lds, stream>>>(x, cout, cws);

  size_t knn_lds = (size_t)16 * DSTRIDE * sizeof(float)                  // dist  ~128 KB
                 + (size_t)16 * 256 * sizeof(float)                      // candD  16 KB
                 + (size_t)16 * 256 * sizeof(int);                       // candI  16 KB
  knn_kernel<<<B_ * (G_ / 16), 256, knn_lds, stream>>>(x, cws, pout, iout);
}